// MambaNode_36060545417604
// MI455X (gfx1250) — compile-verified
//
#include <hip/hip_runtime.h>
#include <math.h>

typedef float v2f __attribute__((ext_vector_type(2)));
typedef float v4f __attribute__((ext_vector_type(4)));
typedef float v8f __attribute__((ext_vector_type(8)));
typedef unsigned int u32x4 __attribute__((ext_vector_type(4)));
typedef int i32x8 __attribute__((ext_vector_type(8)));
typedef int i32x4 __attribute__((ext_vector_type(4)));

#define B_     16
#define L_     4096
#define CDIM   256
#define DMODEL 128
#define DINNER 256
#define DSTATE 16
#define DTRANK 8
#define NTOK   (B_ * L_)

// ---------------- WMMA fp32 GEMM: C[M,N] = A[M,K] @ W[K,N] (+bias) ----------------
// Workgroup tile 128(M) x 64(N), K staged 16 at a time through double-buffered LDS
// filled by the Tensor Data Mover (tensor_load_to_lds, TENSORcnt-tracked).
// 8 waves (wave32): 4x2 wave grid, each wave computes 32x32 via 2x2 v_wmma_f32_16x16x4_f32 tiles.
#define BM 128
#define BN 64
#define KT 16
#define LDA_S 20   // LDS row stride for A tile (floats): 20*r mod 64 distinct -> no bank conflicts
#define LDW_S 80   // LDS row stride for W tile (floats): 80*k mod 64 = 16*k -> disjoint bank groups

// Return workgroup-relative LDS byte offset of a __shared__ pointer
// (generic LDS address carries the offset in its low 32 bits).
__device__ __forceinline__ unsigned lds_off(const void* p)
{
    return (unsigned)(uintptr_t)p;
}

// Issue one TDM 2D tile load: tile_d0 elements (f32) per row, tile_d1 rows,
// row stride stride0 (elements) in global memory; LDS gets pad_amount dwords of
// padding after every pad_interval-coded dwords (reproduces the padded LDS strides).
__device__ __forceinline__ void tdm_load_2d(unsigned lds_addr, const void* gaddr,
                                            unsigned tile_d0, unsigned tile_d1,
                                            unsigned stride0,
                                            unsigned pad_interval_code,
                                            unsigned pad_amount_code)
{
    unsigned long long ga = (unsigned long long)(uintptr_t)gaddr;
    // ---- D# group 0: count=1 | lds_addr<<32 ; global_addr | type(2)<<62 ----
    unsigned long long q0 = 1ull | ((unsigned long long)lds_addr << 32);
    unsigned long long q1 = (ga & ((1ull << 57) - 1)) | (2ull << 62);
    u32x4 g0;
    g0.x = (unsigned)q0; g0.y = (unsigned)(q0 >> 32);
    g0.z = (unsigned)q1; g0.w = (unsigned)(q1 >> 32);
    // ---- D# group 1 ----
    // [15:0] wg_mask=0, [17:16] data_size=2 (4B), [18] atomic_bar=0, [19] iterate=0,
    // [20] pad_enable=1, [24:22] pad_interval, [31:25] pad_amount,
    // [47:32] atomic_bar_addr=0, [79:48] tensor_dim0, [111:80] tensor_dim1,
    // [127:112] tile_dim0, [143:128] tile_dim1, [159:144] tile_dim2=0,
    // [207:160] tensor_dim0_stride, [255:208] tensor_dim1_stride=0 (2D)
    unsigned long long p0 = (2ull << 16) | (1ull << 20)
                          | ((unsigned long long)(pad_interval_code & 7u) << 22)
                          | ((unsigned long long)(pad_amount_code & 127u) << 25)
                          | ((unsigned long long)(tile_d0 & 0xFFFFu) << 48);      // tensor_dim0[15:0] (= tile_d0)
    unsigned long long p1 = ((unsigned long long)(tile_d0 >> 16) & 0xFFFFu)      // tensor_dim0[31:16]
                          | ((unsigned long long)tile_d1 << 16)                   // tensor_dim1 (= tile_d1 rows)
                          | ((unsigned long long)(tile_d0 & 0xFFFFu) << 48);      // tile_dim0
    unsigned long long p2 = ((unsigned long long)(tile_d1 & 0xFFFFu))             // tile_dim1
                          | ((unsigned long long)stride0 << 32);                  // dim0_stride[31:0]
    unsigned long long p3 = ((unsigned long long)stride0 >> 32) & 0xFFFFull;      // dim0_stride[47:32]
    i32x8 g1;
    g1[0] = (int)p0; g1[1] = (int)(p0 >> 32);
    g1[2] = (int)p1; g1[3] = (int)(p1 >> 32);
    g1[4] = (int)p2; g1[5] = (int)(p2 >> 32);
    g1[6] = (int)p3; g1[7] = (int)(p3 >> 32);
    i32x4 z4;
    z4[0] = 0; z4[1] = 0; z4[2] = 0; z4[3] = 0;
    i32x8 z8;
    z8[0] = 0; z8[1] = 0; z8[2] = 0; z8[3] = 0;
    z8[4] = 0; z8[5] = 0; z8[6] = 0; z8[7] = 0;
    // 6-arg toolchain form: (g0, g1, g2, g3, extra, cpol)
    __builtin_amdgcn_tensor_load_to_lds(g0, g1, z4, z4, z8, 0);
}

__global__ __launch_bounds__(256)
void gemm_wmma_f32(const float* __restrict__ A0, const float* __restrict__ A1,
                   int ksplit,                 // k < ksplit from A0 (row stride ksplit), else A1 (row stride K-ksplit)
                   const float* __restrict__ W, const float* __restrict__ bias,
                   float* __restrict__ Cc, int M, int N, int K, int ldc)
{
    __shared__ float sA[2][BM * LDA_S];
    __shared__ float sW[2][KT * LDW_S];

    const int tid  = threadIdx.x;
    const int lane = tid & 31;
    const int m0   = blockIdx.x * BM;
    const int n0   = blockIdx.y * BN;
    const int wave = tid >> 5;
    const int wm   = (wave & 3) * 32;   // wave M offset inside block tile
    const int wn   = (wave >> 2) * 32;  // wave N offset inside block tile

    v8f acc[2][2];
    for (int i = 0; i < 2; ++i)
        for (int j = 0; j < 2; ++j)
            for (int e = 0; e < 8; ++e) acc[i][j][e] = 0.0f;

    const int nkt = K / KT;
    const bool issuer = (wave == 0);    // wave-uniform: only wave 0 drives the TDM

    // TDM tile issue: A tile = KT x BM rows (stride = A row stride), W tile = BN x KT rows (stride N)
    auto issue = [&](int kt, int buf) {
        const int k0 = kt * KT;
        const float* abase;
        int astride;
        if (k0 < ksplit) { abase = A0 + (size_t)m0 * ksplit + k0;                 astride = ksplit; }
        else             { abase = A1 + (size_t)m0 * (K - ksplit) + (k0 - ksplit); astride = K - ksplit; }
        // A: rows of 16 dwords, pad 4 dwords -> LDS stride 20  (interval code 3 = 16 dw, amount code 3 = 4 dw)
        tdm_load_2d(lds_off(&sA[buf][0]), abase, KT, BM, (unsigned)astride, 3u, 3u);
        // W: rows of 64 dwords, pad 16 dwords -> LDS stride 80 (interval code 5 = 64 dw, amount code 15 = 16 dw)
        tdm_load_2d(lds_off(&sW[buf][0]), W + (size_t)k0 * N + n0, BN, KT, (unsigned)N, 5u, 15u);
    };

    if (issuer) issue(0, 0);

    for (int kt = 0; kt < nkt; ++kt) {
        const int buf = kt & 1;
        if (issuer) {
            if (kt + 1 < nkt) issue(kt + 1, buf ^ 1);
            // TDM ops from one wave complete in order: <=2 outstanding means the
            // current tile's pair (A,W) has landed in LDS.
            if (kt + 1 < nkt) __builtin_amdgcn_s_wait_tensorcnt((short)2);
            else              __builtin_amdgcn_s_wait_tensorcnt((short)0);
        }
        __syncthreads();   // release compute waves once wave0 observed completion

        const float* As = &sA[buf][0];
        const float* Ws = &sW[buf][0];
        #pragma unroll
        for (int kk = 0; kk < KT; kk += 4) {
            const int klane = kk + ((lane >> 4) << 1);   // K=0/1 for lanes 0-15, K=2/3 for lanes 16-31
            #pragma unroll
            for (int mt = 0; mt < 2; ++mt) {
                const int mrow = wm + mt * 16 + (lane & 15);
                v2f a = *(const v2f*)&As[mrow * LDA_S + klane];
                #pragma unroll
                for (int nt = 0; nt < 2; ++nt) {
                    const int ncol = wn + nt * 16 + (lane & 15);
                    v2f b;
                    b.x = Ws[klane * LDW_S + ncol];
                    b.y = Ws[(klane + 1) * LDW_S + ncol];
                    acc[mt][nt] = __builtin_amdgcn_wmma_f32_16x16x4_f32(
                        false, a, false, b, (short)0, acc[mt][nt], false, false);
                }
            }
        }
        __syncthreads();   // all reads of buf done before TDM overwrites it next iter
    }

    // Epilogue: C/D layout — VGPR r holds M=r (lanes 0-15) / M=8+r (lanes 16-31), N = lane%16.
    const int colg  = lane & 15;
    const int rbase = (lane >> 4) * 8;
    #pragma unroll
    for (int mt = 0; mt < 2; ++mt)
        #pragma unroll
        for (int nt = 0; nt < 2; ++nt) {
            const int col = n0 + wn + nt * 16 + colg;
            const float bv = bias ? bias[col] : 0.0f;
            #pragma unroll
            for (int r = 0; r < 8; ++r) {
                const int row = m0 + wm + mt * 16 + rbase + r;
                Cc[(size_t)row * ldc + col] = acc[mt][nt][r] + bv;
            }
        }
}

// ---------------- W_o = W_op @ W_out, b_o = b_op @ W_out + b_out ----------------
__global__ __launch_bounds__(256)
void fuse_ow(const float* __restrict__ Wop, const float* __restrict__ bop,
             const float* __restrict__ Wout, const float* __restrict__ bout,
             float* __restrict__ Wo, float* __restrict__ bo)
{
    int j = threadIdx.x;   // 0..255 (output column)
    int i = blockIdx.x;    // 0..255 rows, block 256 does the bias
    if (i < 256) {
        float s = 0.f;
        for (int m = 0; m < 128; ++m) s += Wop[i * 128 + m] * Wout[m * 256 + j];
        Wo[i * 256 + j] = s;
    } else {
        float s = bout[j];
        for (int m = 0; m < 128; ++m) s += bop[m] * Wout[m * 256 + j];
        bo[j] = s;
    }
}

// ---------------- pad W_xproj (256x40) -> (256x64) ----------------
__global__ __launch_bounds__(256)
void pack_xproj(const float* __restrict__ Wx, float* __restrict__ Wp)
{
    int idx = blockIdx.x * 256 + threadIdx.x;  // 256*64
    int k = idx >> 6, n = idx & 63;
    Wp[idx] = (n < 40) ? Wx[k * 40 + n] : 0.f;
}

// ---------------- depthwise causal conv (K=4) + SiLU ----------------
__global__ __launch_bounds__(256)
void conv_silu(const float* __restrict__ xz, const float* __restrict__ cw,
               const float* __restrict__ cb, float* __restrict__ xcs)
{
    int idx = blockIdx.x * 256 + threadIdx.x;     // over NTOK*256
    int d   = idx & 255;
    int tok = idx >> 8;
    int t   = tok & (L_ - 1);
    float acc = cb[d];
    #pragma unroll
    for (int k = 0; k < 4; ++k) {
        int tt = t - 3 + k;
        if (tt >= 0) acc += xz[(size_t)(tok - 3 + k) * 512 + d] * cw[d * 4 + k];
    }
    xcs[idx] = acc / (1.f + __expf(-acc));        // silu
}

// ---------------- dt = softplus(dbl[:, :8] @ W_dt + b_dt) ----------------
__global__ __launch_bounds__(256)
void dt_softplus(const float* __restrict__ dbl, const float* __restrict__ Wdt,
                 const float* __restrict__ bdt, float* __restrict__ dtb)
{
    int idx = blockIdx.x * 256 + threadIdx.x;     // over NTOK*256
    int d   = idx & 255;
    size_t tok = (size_t)(idx >> 8);
    float acc = bdt[d];
    #pragma unroll
    for (int r = 0; r < 8; ++r) acc += dbl[tok * 64 + r] * Wdt[r * 256 + d];
    dtb[idx] = (acc > 20.f) ? acc : __logf(1.f + __expf(acc));
}

// ---------------- selective scan: 1 block per batch, 1 lane per channel ----------------
#define TCH 64
__global__ __launch_bounds__(256)
void scan_kernel(const float* __restrict__ dbl, const float* dtb,
                 const float* __restrict__ xcs, const float* __restrict__ xz,
                 const float* __restrict__ Alog, const float* __restrict__ Dp,
                 float* yb /* may alias dtb: each element read-then-written by same thread */)
{
    __shared__ float sbc[TCH * 32];               // per-token [B(16) | C(16)]
    const int d = threadIdx.x;
    const int b = blockIdx.x;

    float a[DSTATE], h[DSTATE];
    #pragma unroll
    for (int s = 0; s < DSTATE; ++s) { a[s] = -__expf(Alog[d * DSTATE + s]); h[s] = 0.f; }
    const float dpar = Dp[d];

    for (int t0 = 0; t0 < L_; t0 += TCH) {
        // prefetch next chunk's streaming operands while this chunk computes
        if (t0 + TCH < L_) {
            const size_t ntok = (size_t)b * L_ + t0 + TCH;
            __builtin_prefetch(&dtb[ntok * 256 + d], 0, 1);
            __builtin_prefetch(&xcs[ntok * 256 + d], 0, 1);
            __builtin_prefetch(&xz[ntok * 512 + 256 + d], 0, 1);
            __builtin_prefetch(&dbl[ntok * 64 + (d & 31)], 0, 1);
        }
        __syncthreads();
        #pragma unroll
        for (int it = 0; it < 8; ++it) {          // 64 tokens * 32 floats / 256 threads
            int q  = threadIdx.x + it * 256;
            int tt = q >> 5, j = q & 31;
            sbc[q] = dbl[(size_t)(b * L_ + t0 + tt) * 64 + 8 + j];
        }
        __syncthreads();

        for (int tt = 0; tt < TCH; ++tt) {
            const size_t tok = (size_t)b * L_ + t0 + tt;
            const float dtv = dtb[tok * 256 + d];
            const float xv  = xcs[tok * 256 + d];
            const float zv  = xz[tok * 512 + 256 + d];
            const float dbx = dtv * xv;
            const float* bc = &sbc[tt * 32];
            float y = 0.f;
            #pragma unroll
            for (int s = 0; s < DSTATE; ++s) {
                float e = __expf(dtv * a[s]);
                h[s] = e * h[s] + bc[s] * dbx;
                y += h[s] * bc[16 + s];
            }
            y = y + xv * dpar;
            const float sz = zv / (1.f + __expf(-zv));
            yb[tok * 256 + d] = y * sz;
        }
    }
}

// ---------------- host-side launcher ----------------
extern "C" void kernel_launch(void* const* d_in, const int* in_sizes, int n_in,
                              void* d_out, int out_size, void* d_ws, size_t ws_size,
                              hipStream_t stream)
{
    const float* x      = (const float*)d_in[0];
    const float* qk     = (const float*)d_in[1];
    const float* W_mix  = (const float*)d_in[2];
    const float* b_mix  = (const float*)d_in[3];
    const float* W_in   = (const float*)d_in[4];
    const float* b_in   = (const float*)d_in[5];
    const float* conv_w = (const float*)d_in[6];
    const float* conv_b = (const float*)d_in[7];
    const float* W_xprj = (const float*)d_in[8];
    const float* W_dt   = (const float*)d_in[9];
    const float* b_dt   = (const float*)d_in[10];
    const float* A_log  = (const float*)d_in[11];
    const float* D_par  = (const float*)d_in[12];
    const float* W_op   = (const float*)d_in[13];
    const float* b_op   = (const float*)d_in[14];
    const float* W_out  = (const float*)d_in[15];
    const float* b_out  = (const float*)d_in[16];

    float* ws   = (float*)d_ws;
    float* hbuf = ws;                                   // NTOK*128
    float* xzb  = hbuf + (size_t)NTOK * 128;            // NTOK*512
    float* xcs  = xzb  + (size_t)NTOK * 512;            // NTOK*256
    float* dbl  = xcs  + (size_t)NTOK * 256;            // NTOK*64 (padded xproj out)
    float* dtb  = dbl  + (size_t)NTOK * 64;             // NTOK*256 (dt, then gated y in-place)
    float* Wo   = dtb  + (size_t)NTOK * 256;            // 256*256
    float* bo   = Wo   + 256 * 256;                     // 256
    float* Wxp  = bo   + 256;                           // 256*64

    fuse_ow   <<<257, 256, 0, stream>>>(W_op, b_op, W_out, b_out, Wo, bo);
    pack_xproj<<<64, 256, 0, stream>>>(W_xprj, Wxp);

    // h = [x|qk] @ W_mix + b_mix       (M=65536, N=128, K=512; A concat handled in TDM issue)
    gemm_wmma_f32<<<dim3(NTOK / BM, 128 / BN), 256, 0, stream>>>(
        x, qk, 256, W_mix, b_mix, hbuf, NTOK, 128, 512, 128);

    // xz = h @ W_in + b_in             (M=65536, N=512, K=128)
    gemm_wmma_f32<<<dim3(NTOK / BM, 512 / BN), 256, 0, stream>>>(
        hbuf, nullptr, 128, W_in, b_in, xzb, NTOK, 512, 128, 512);

    conv_silu<<<NTOK, 256, 0, stream>>>(xzb, conv_w, conv_b, xcs);

    // dbl = xcs @ W_xproj(padded to 64) (M=65536, N=64, K=256)
    gemm_wmma_f32<<<dim3(NTOK / BM, 1), 256, 0, stream>>>(
        xcs, nullptr, 256, Wxp, nullptr, dbl, NTOK, 64, 256, 64);

    dt_softplus<<<NTOK, 256, 0, stream>>>(dbl, W_dt, b_dt, dtb);

    // scan + gating; gated y overwrites dtb in place (safe: per-element read-then-write by same thread)
    scan_kernel<<<B_, 256, 0, stream>>>(dbl, dtb, xcs, xzb, A_log, D_par, dtb);

    // out = y @ (W_op@W_out) + (b_op@W_out + b_out)   (M=65536, N=256, K=256)
    gemm_wmma_f32<<<dim3(NTOK / BM, 256 / BN), 256, 0, stream>>>(
        dtb, nullptr, 256, Wo, bo, (float*)d_out, NTOK, 256, 256, 256);
}